// SHEquivariantDecoder_76003741270185
// MI455X (gfx1250) — compile-verified
//
#include <hip/hip_runtime.h>
#include <hip/hip_bf16.h>
#include <math.h>

typedef __attribute__((ext_vector_type(2))) float v2f;
typedef __attribute__((ext_vector_type(8))) float v8f;
typedef __attribute__((address_space(3))) float lds_float;

#define NBATCH 256
#define NLATENT 256
#define INDIM 800
#define ROWLEN 235520
#define EPSV 1e-5f

// workspace layout (floats)
#define OFF_FEATS 0
#define OFF_XW    204800
#define OFF_INV   409600
#define OFF_R     409856
#define OFF_MW    411136
#define OFF_T     411392

__constant__ int cFOFF[5] = {0, 32, 128, 288, 512};          // elem offset of l-block in 800-row
__constant__ int cD3[5]   = {1, 3, 5, 7, 9};
__constant__ int cCL[5]   = {5120, 8192, 10240, 10240, 9216}; // channels per l
__constant__ int cCHB[5]  = {0, 5120, 13312, 23552, 33792};   // channel base per l
__constant__ int cELB[5]  = {0, 5120, 29696, 80896, 152576};  // element base per l in 235520-row

// 42 triples (l1,l2,l3) in reference order + position within contrib[l3]
__constant__ int cT1[42] = {0,0,0,0,0, 1,1,1, 1,1,1, 1,1,1, 1,1, 2,2,2,2,2, 2,2,2,2, 2,2,2, 3,3,3,3,3, 3,3,3,3, 4,4,4,4,4};
__constant__ int cT2[42] = {0,1,2,3,4, 1,1,1, 2,2,2, 3,3,3, 4,4, 2,2,2,2,2, 3,3,3,3, 4,4,4, 3,3,3,3,3, 4,4,4,4, 4,4,4,4,4};
__constant__ int cT3[42] = {0,1,2,3,4, 0,1,2, 1,2,3, 2,3,4, 3,4, 0,1,2,3,4, 1,2,3,4, 2,3,4, 0,1,2,3,4, 1,2,3,4, 0,1,2,3,4};
__constant__ int cTP[42] = {0,0,0,0,0, 1,1,1, 2,2,1, 3,2,1, 3,2, 2,3,4,4,3, 4,5,5,4, 6,6,5, 3,5,7,7,6, 6,8,8,7, 4,7,9,9,8};

static __device__ __forceinline__ v8f wmma_f32(v2f a, v2f b, v8f c) {
  return __builtin_amdgcn_wmma_f32_16x16x4_f32(false, a, false, b, (short)0, c, false, false);
}

static __device__ __forceinline__ unsigned lds_offset(const float* p) {
  return (unsigned)(size_t)(lds_float*)p;
}

// ---------------- Kernel 1: feats = z @ fc_w + fc_b  (256 x 800, K=256) ----------------
__global__ __launch_bounds__(32) void k_fc(const float* __restrict__ z, const float* __restrict__ fcw,
                                           const float* __restrict__ fcb, float* __restrict__ feats) {
  int n0 = blockIdx.x * 16, m0 = blockIdx.y * 16;
  int lane = threadIdx.x, l15 = lane & 15;
  int khalf = (lane < 16) ? 0 : 2;
  int rhalf = (lane < 16) ? 0 : 8;
  int arow = m0 + l15;
  v8f acc = {};
  for (int ks = 0; ks < 64; ++ks) {
    int kb = ks * 4 + khalf;
    v2f a, b;
    a.x = z[arow * NLATENT + kb];
    a.y = z[arow * NLATENT + kb + 1];
    b.x = fcw[kb * INDIM + n0 + l15];
    b.y = fcw[(kb + 1) * INDIM + n0 + l15];
    acc = wmma_f32(a, b, acc);
  }
  int col = n0 + l15;
  float bias = fcb[col];
  for (int r = 0; r < 8; ++r)
    feats[(m0 + r + rhalf) * INDIM + col] = acc[r] + bias;
}

// ---------------- Kernel 2: per-l channel mix  xw[b,j,m] = sum_i feats[b,i,m] linw[l][i,j] ----------------
__global__ __launch_bounds__(32) void k_lin(const float* __restrict__ feats, const float* __restrict__ linw,
                                            const float* __restrict__ linb, float* __restrict__ xw) {
  int j0 = blockIdx.x * 16, row0 = blockIdx.y * 16;
  int lane = threadIdx.x, l15 = lane & 15;
  int khalf = (lane < 16) ? 0 : 2;
  int rhalf = (lane < 16) ? 0 : 8;
  int l = (row0 < 256) ? 0 : (row0 < 1024) ? 1 : (row0 < 2304) ? 2 : (row0 < 4096) ? 3 : 4;
  int d = 2 * l + 1;
  int base = (l == 0) ? 0 : (l == 1) ? 256 : (l == 2) ? 1024 : (l == 3) ? 2304 : 4096;
  int foff = cFOFF[l];
  int rowA = row0 + l15 - base;
  int bA = rowA / d, mA = rowA % d;
  v8f acc = {};
  for (int ks = 0; ks < 8; ++ks) {
    int i0 = ks * 4 + khalf;
    v2f a, b;
    a.x = feats[bA * INDIM + foff + i0 * d + mA];
    a.y = feats[bA * INDIM + foff + (i0 + 1) * d + mA];
    b.x = linw[l * 1024 + i0 * 32 + j0 + l15];
    b.y = linw[l * 1024 + (i0 + 1) * 32 + j0 + l15];
    acc = wmma_f32(a, b, acc);
  }
  float bias = (l == 0) ? linb[j0 + l15] : 0.f;
  for (int r = 0; r < 8; ++r) {
    int row = row0 + r + rhalf - base;
    int bb = row / d, mm = row % d;
    xw[bb * INDIM + foff + (j0 + l15) * d + mm] = acc[r] + bias;
  }
}

// ---------------- Kernel 3: triple products, write unnormalized y into d_out ----------------
// One block = one triple t, 8 batches (one per wave). w3j[t] staged to LDS via async
// copy (ASYNCcnt); x1/x2 tiles staged per wave. Y[j,(i,k)] = X2(32 x d2) * P(d2 x 32*d3),
// P[n,(i,k)] = sum_m x1[i,m] w[m,n,k]; each B-fragment feeds two WMMAs (both M-tiles).
__global__ __launch_bounds__(256) void k_tp(const float* __restrict__ xw, const float* __restrict__ w3j,
                                            float* __restrict__ out) {
  __shared__ float sw[729];
  __shared__ float sx1[2304];
  __shared__ float sx2[2304];
  int t = blockIdx.x;
  int tid = threadIdx.x;
  int wave = tid >> 5, lane = tid & 31;
  int b = blockIdx.y * 8 + wave;
  int l15 = lane & 15;
  int khalf = (lane < 16) ? 0 : 2;
  int rhalf = (lane < 16) ? 0 : 8;
  int l1 = cT1[t], l2 = cT2[t], l3 = cT3[t], pos = cTP[t];
  int d1 = 2 * l1 + 1, d2 = 2 * l2 + 1, d3 = 2 * l3 + 1;

  // async-stage w3j[t] (729 floats) into LDS
  const float* wsrc = w3j + t * 729;
  for (int idx = tid; idx < 729; idx += 256) {
    unsigned lofs = lds_offset(&sw[idx]);
    unsigned goff = (unsigned)(idx * 4);
    asm volatile("global_load_async_to_lds_b32 %0, %1, %2"
                 :: "v"(lofs), "v"(goff), "s"(wsrc)
                 : "memory");
  }

  // stage this wave's x1/x2 tiles (normal LDS stores; lane == channel row)
  const float* x1 = xw + b * INDIM + cFOFF[l1];
  const float* x2 = xw + b * INDIM + cFOFF[l2];
  int xo = wave * 288;
  for (int m = 0; m < d1; ++m) sx1[xo + lane * d1 + m] = x1[lane * d1 + m];
  for (int n = 0; n < d2; ++n) sx2[xo + lane * d2 + n] = x2[lane * d2 + n];

  asm volatile("s_wait_asynccnt 0x0" ::: "memory");
  __syncthreads();

  int outBase = b * ROWLEN + cELB[l3] + pos * 1024 * d3;
  int Ksteps = (d2 + 3) >> 2;
  for (int nt = 0; nt < 2 * d3; ++nt) {
    int col = nt * 16 + l15;
    int ic = col / d3, kk = col % d3;
    v8f acc0 = {}, acc1 = {};
    for (int ks = 0; ks < Ksteps; ++ks) {
      int nb = ks * 4 + khalf;
      v2f bf = {0.f, 0.f};
      v2f a0 = {0.f, 0.f}, a1 = {0.f, 0.f};
      if (nb < d2) {
        float s = 0.f;
        for (int m = 0; m < d1; ++m) s += sx1[xo + ic * d1 + m] * sw[m * 81 + nb * 9 + kk];
        bf.x = s;
        a0.x = sx2[xo + l15 * d2 + nb];
        a1.x = sx2[xo + (16 + l15) * d2 + nb];
      }
      if (nb + 1 < d2) {
        float s = 0.f;
        for (int m = 0; m < d1; ++m) s += sx1[xo + ic * d1 + m] * sw[m * 81 + (nb + 1) * 9 + kk];
        bf.y = s;
        a0.y = sx2[xo + l15 * d2 + nb + 1];
        a1.y = sx2[xo + (16 + l15) * d2 + nb + 1];
      }
      acc0 = wmma_f32(a0, bf, acc0);  // M-tile j=0..15
      acc1 = wmma_f32(a1, bf, acc1);  // M-tile j=16..31
    }
    int cb = outBase + ic * 32 * d3 + kk;
    for (int r = 0; r < 8; ++r) {
      out[cb + (r + rhalf) * d3]      = acc0[r];
      out[cb + (16 + r + rhalf) * d3] = acc1[r];
    }
  }
}

// ---------------- Kernel 4: per-batch stats -> inv_b, r_{b,l}, meanW_b ----------------
__global__ __launch_bounds__(256) void k_stats(const float* __restrict__ out, float* __restrict__ ws) {
  int b = blockIdx.x, tid = threadIdx.x;
  __shared__ float red[256];
  float accP[5] = {0.f, 0.f, 0.f, 0.f, 0.f};
  float accW = 0.f;
  const float* row = out + b * ROWLEN;
  for (int l = 0; l < 5; ++l) {
    int d3 = cD3[l], cl = cCL[l];
    const float* base = row + cELB[l];
    for (int c = tid; c < cl; c += 256) {
      const float* p = base + c * d3;
      float q = 0.f;
      for (int k = 0; k < d3; ++k) { float v = p[k]; q += v * v; }
      float n = sqrtf(q);
      float pc = q / ((n + EPSV) * (n + EPSV));
      accP[l] += pc;
      if (l == 0) accW += p[0] / (n + EPSV);
    }
  }
  float sums[6];
  for (int s = 0; s < 6; ++s) {
    red[tid] = (s < 5) ? accP[s] : accW;
    __syncthreads();
    for (int off = 128; off > 0; off >>= 1) {
      if (tid < off) red[tid] += red[tid + off];
      __syncthreads();
    }
    sums[s] = red[0];
    __syncthreads();
  }
  if (tid == 0) {
    float power = sums[0] + sums[1] + sums[2] + sums[3] + sums[4];
    float inv = 1.f / (sqrtf(power / 235520.f) + EPSV);
    float meanW = inv * sums[5] / 5120.f;
    float ms0 = inv * inv * sums[0] / 5120.f - meanW * meanW;
    if (ms0 < 0.f) ms0 = 0.f;
    ws[OFF_INV + b] = inv;
    ws[OFF_MW + b] = meanW;
    ws[OFF_R + b * 5 + 0] = 1.f / (sqrtf(ms0) + EPSV);
    for (int l = 1; l < 5; ++l) {
      float ms = inv * inv * sums[l] / (float)(cCL[l] * cD3[l]);
      ws[OFF_R + b * 5 + l] = 1.f / (sqrtf(ms) + EPSV);
    }
  }
}

// ---------------- Kernel 5: per-channel-over-batch stats (l>=1) -> t_{l,c} ----------------
__global__ __launch_bounds__(256) void k_chanstats(const float* __restrict__ out, float* __restrict__ ws) {
  __shared__ float sInv[256];
  __shared__ float sR[1280];
  int tid = threadIdx.x;
  sInv[tid] = ws[OFF_INV + tid];
  for (int i = tid; i < 1280; i += 256) sR[i] = ws[OFF_R + i];
  __syncthreads();
  int c = 5120 + blockIdx.x * 256 + tid;
  int l = (c < 13312) ? 1 : (c < 23552) ? 2 : (c < 33792) ? 3 : 4;
  int d3 = cD3[l];
  int elem = cELB[l] + (c - cCHB[l]) * d3;
  float acc = 0.f;
  for (int b = 0; b < 256; ++b) {
    const float* p = out + b * ROWLEN + elem;
    float q = 0.f;
    for (int k = 0; k < d3; ++k) { float v = p[k]; q += v * v; }
    float n = sqrtf(q);
    float pc = q / ((n + EPSV) * (n + EPSV));
    float ir = sInv[b] * sR[b * 5 + l];
    acc += pc * ir * ir;
  }
  ws[OFF_T + c] = 1.f / (sqrtf(acc / (256.f * (float)d3)) + EPSV);
}

// ---------------- Kernel 6a: l=0 elementwise through step 4 ----------------
__global__ __launch_bounds__(256) void k_l0a(float* __restrict__ out, const float* __restrict__ ws) {
  int b = blockIdx.y;
  int c = blockIdx.x * 256 + threadIdx.x;
  float inv = ws[OFF_INV + b], mw = ws[OFF_MW + b], r0 = ws[OFF_R + b * 5];
  float* p = out + b * ROWLEN + c;
  float y = *p;
  float w = y / (fabsf(y) + EPSV) * inv;
  *p = (w - mw) * r0;
}

// ---------------- Kernel 6b: l=0 batch-centering + per-channel batch RMS ----------------
__global__ __launch_bounds__(256) void k_l0b(float* __restrict__ out) {
  int c = blockIdx.x * 256 + threadIdx.x;
  float s = 0.f, ss = 0.f;
  for (int b = 0; b < 256; ++b) {
    float v = out[b * ROWLEN + c];
    s += v; ss += v * v;
  }
  float mean = s / 256.f;
  float var = ss / 256.f - mean * mean;
  if (var < 0.f) var = 0.f;
  float sc = 1.f / (sqrtf(var) + EPSV);
  for (int b = 0; b < 256; ++b) {
    float v = out[b * ROWLEN + c];
    out[b * ROWLEN + c] = (v - mean) * sc;
  }
}

// ---------------- Kernel 7: final in-place scale for l>=1 ----------------
__global__ __launch_bounds__(256) void k_scale(float* __restrict__ out, const float* __restrict__ ws) {
  int b = blockIdx.y;
  int c = 5120 + blockIdx.x * 256 + threadIdx.x;
  int l = (c < 13312) ? 1 : (c < 23552) ? 2 : (c < 33792) ? 3 : 4;
  int d3 = cD3[l];
  float* p = out + b * ROWLEN + cELB[l] + (c - cCHB[l]) * d3;
  float q = 0.f;
  for (int k = 0; k < d3; ++k) { float v = p[k]; q += v * v; }
  float n = sqrtf(q);
  float S = ws[OFF_INV + b] * ws[OFF_R + b * 5 + l] * ws[OFF_T + c] / (n + EPSV);
  for (int k = 0; k < d3; ++k) p[k] *= S;
}

extern "C" void kernel_launch(void* const* d_in, const int* in_sizes, int n_in,
                              void* d_out, int out_size, void* d_ws, size_t ws_size,
                              hipStream_t stream) {
  const float* z    = (const float*)d_in[0];
  const float* fcw  = (const float*)d_in[1];
  const float* fcb  = (const float*)d_in[2];
  const float* linw = (const float*)d_in[3];
  const float* linb = (const float*)d_in[4];
  const float* w3j  = (const float*)d_in[5];
  float* out = (float*)d_out;
  float* ws  = (float*)d_ws;
  float* feats = ws + OFF_FEATS;
  float* xw    = ws + OFF_XW;

  k_fc<<<dim3(50, 16), 32, 0, stream>>>(z, fcw, fcb, feats);
  k_lin<<<dim3(2, 400), 32, 0, stream>>>(feats, linw, linb, xw);
  k_tp<<<dim3(42, 32), 256, 0, stream>>>(xw, w3j, out);
  k_stats<<<256, 256, 0, stream>>>(out, ws);
  k_chanstats<<<148, 256, 0, stream>>>(out, ws);
  k_l0a<<<dim3(20, 256), 256, 0, stream>>>(out, ws);
  k_l0b<<<20, 256, 0, stream>>>(out);
  k_scale<<<dim3(148, 256), 256, 0, stream>>>(out, ws);
}